// MakespanLoss_88742614270084
// MI455X (gfx1250) — compile-verified
//
#include <hip/hip_runtime.h>
#include <math.h>

typedef __attribute__((ext_vector_type(2))) float v2f;
typedef __attribute__((ext_vector_type(4))) float v4f;
typedef __attribute__((ext_vector_type(8))) float v8f;

#define B_   64
#define N_   1024
#define P_   256
#define TOT  (B_ * N_ * P_)

// d_ws layout (floats)
#define NNORMS      (B_ * N_)                         // 65536 row norms
#define BCE_BLOCKS  4096
#define ST          32                                // 32x32 super-tiles per dim
#define NSUPER      (ST * (ST + 1) / 2)               // 528 upper-tri super-tiles
#define DIST_BLOCKS (B_ * NSUPER)                     // 33792
#define BCE_OFF     (NNORMS)
#define DIST_OFF    (BCE_OFF + BCE_BLOCKS)

// ---------------------------------------------------------------------------
// Kernel 1: row norms  sq[b*N+n] = sum_p x[b,n,p]^2   (one wave per row)
// ---------------------------------------------------------------------------
__global__ void mk_norms(const float* __restrict__ x, float* __restrict__ norms) {
    int wave = (blockIdx.x * blockDim.x + threadIdx.x) >> 5;   // 65536 waves
    int lane = threadIdx.x & 31;
    const float* row = x + (size_t)wave * P_;
    float s = 0.f;
#pragma unroll
    for (int i = 0; i < 2; ++i) {                // 256 floats = 32 lanes * 2 * float4
        v4f v = *(const v4f*)(row + lane * 4 + i * 128);
        s += v.x * v.x + v.y * v.y + v.z * v.z + v.w * v.w;
    }
#pragma unroll
    for (int off = 16; off; off >>= 1) s += __shfl_xor(s, off, 32);
    if (lane == 0) norms[wave] = s;
}

// ---------------------------------------------------------------------------
// Kernel 2: BCE partial sums (stable logaddexp form), fixed-slot partials
// ---------------------------------------------------------------------------
__global__ void mk_bce(const float* __restrict__ x, const float* __restrict__ t,
                       float* __restrict__ part) {
    const int nvec = TOT / 4;
    float s = 0.f;
    for (int i = blockIdx.x * blockDim.x + threadIdx.x; i < nvec;
         i += gridDim.x * blockDim.x) {
        v4f xv = ((const v4f*)x)[i];
        v4f tv = ((const v4f*)t)[i];
#pragma unroll
        for (int c = 0; c < 4; ++c) {
            float xx = xv[c], tt = tv[c];
            s += fmaxf(xx, 0.f) - xx * tt + log1pf(__expf(-fabsf(xx)));
        }
    }
    __shared__ float red[256];
    red[threadIdx.x] = s;
    __syncthreads();
#pragma unroll
    for (int o = 128; o; o >>= 1) {
        if ((int)threadIdx.x < o) red[threadIdx.x] += red[threadIdx.x + o];
        __syncthreads();
    }
    if (threadIdx.x == 0) part[blockIdx.x] = red[0];
}

// ---------------------------------------------------------------------------
// Kernel 3: pairwise-distance tiles via V_WMMA_F32_16X16X4_F32.
// One wave computes a 32x32 gram super-tile (2x2 WMMA accumulator tiles),
// then the sqrt-distance epilogue and a wave reduction.
// ---------------------------------------------------------------------------
__global__ void __launch_bounds__(32)
mk_dist(const float* __restrict__ x, const float* __restrict__ norms,
        float* __restrict__ part) {
    const int bid = blockIdx.x;
    const int b   = bid / NSUPER;
    int s         = bid % NSUPER;

    // decode upper-triangular super-tile (ti <= tj); wave-uniform scalar loop
    int ti = 0;
    while (s >= ST - ti) { s -= ST - ti; ++ti; }
    const int tj = ti + s;
    const int i0 = ti * 32, j0 = tj * 32;

    const float* X  = x + (size_t)b * N_ * P_;
    const float* nb = norms + b * N_;

    const int lane = threadIdx.x;
    const int r    = lane & 15;      // row within fragment
    const int h    = lane >> 4;      // K half: lanes 0-15 -> K+0, 16-31 -> K+2

    const float* arow0 = X + (size_t)(i0 + r) * P_ + 2 * h;
    const float* arow1 = arow0 + 16 * (size_t)P_;
    const float* brow0 = X + (size_t)(j0 + r) * P_ + 2 * h;
    const float* brow1 = brow0 + 16 * (size_t)P_;

    v8f acc00 = {}, acc01 = {}, acc10 = {}, acc11 = {};

#pragma unroll 4
    for (int k = 0; k < P_; k += 4) {
        v2f a0 = *(const v2f*)(arow0 + k);
        v2f a1 = *(const v2f*)(arow1 + k);
        v2f b0 = *(const v2f*)(brow0 + k);
        v2f b1 = *(const v2f*)(brow1 + k);
        acc00 = __builtin_amdgcn_wmma_f32_16x16x4_f32(false, a0, false, b0,
                                                      (short)0, acc00, false, false);
        acc01 = __builtin_amdgcn_wmma_f32_16x16x4_f32(false, a0, false, b1,
                                                      (short)0, acc01, false, false);
        acc10 = __builtin_amdgcn_wmma_f32_16x16x4_f32(false, a1, false, b0,
                                                      (short)0, acc10, false, false);
        acc11 = __builtin_amdgcn_wmma_f32_16x16x4_f32(false, a1, false, b1,
                                                      (short)0, acc11, false, false);
    }

    // epilogue: C/D layout -> element (M = 8*h + v, N = r) within each 16x16 tile
    const float sqn0 = nb[j0 + r];
    const float sqn1 = nb[j0 + 16 + r];
    const int   n0   = j0 + r;
    const int   n1   = n0 + 16;

    float sum = 0.f;
#pragma unroll
    for (int v = 0; v < 8; ++v) {
        const int   m0   = i0 + 8 * h + v;
        const int   m1   = m0 + 16;
        const float sqm0 = nb[m0];
        const float sqm1 = nb[m1];
        if (m0 < n0) sum += sqrtf(fmaxf(sqm0 + sqn0 - 2.f * acc00[v], 0.f));
        if (m0 < n1) sum += sqrtf(fmaxf(sqm0 + sqn1 - 2.f * acc01[v], 0.f));
        if (m1 < n0) sum += sqrtf(fmaxf(sqm1 + sqn0 - 2.f * acc10[v], 0.f));
        if (m1 < n1) sum += sqrtf(fmaxf(sqm1 + sqn1 - 2.f * acc11[v], 0.f));
    }
#pragma unroll
    for (int off = 16; off; off >>= 1) sum += __shfl_xor(sum, off, 32);
    if (lane == 0) part[bid] = sum;
}

// ---------------------------------------------------------------------------
// Kernel 4: deterministic final reduction -> scalar
// result = bce_total / (B*N*P)  -  dist_total / N
// ---------------------------------------------------------------------------
__global__ void mk_finish(const float* __restrict__ bceP,
                          const float* __restrict__ distP,
                          float* __restrict__ out) {
    __shared__ float rb[256], rd[256];
    float sb = 0.f, sd = 0.f;
    for (int i = threadIdx.x; i < BCE_BLOCKS; i += 256) sb += bceP[i];
    for (int i = threadIdx.x; i < DIST_BLOCKS; i += 256) sd += distP[i];
    rb[threadIdx.x] = sb;
    rd[threadIdx.x] = sd;
    __syncthreads();
#pragma unroll
    for (int o = 128; o; o >>= 1) {
        if ((int)threadIdx.x < o) {
            rb[threadIdx.x] += rb[threadIdx.x + o];
            rd[threadIdx.x] += rd[threadIdx.x + o];
        }
        __syncthreads();
    }
    if (threadIdx.x == 0)
        out[0] = rb[0] / (float)TOT - rd[0] / (float)N_;
}

// ---------------------------------------------------------------------------
extern "C" void kernel_launch(void* const* d_in, const int* in_sizes, int n_in,
                              void* d_out, int out_size, void* d_ws, size_t ws_size,
                              hipStream_t stream) {
    const float* x = (const float*)d_in[0];   // output (logits)
    const float* t = (const float*)d_in[1];   // target_priorities
    float* out     = (float*)d_out;
    float* ws      = (float*)d_ws;

    float* norms = ws;             // 65536 floats
    float* bceP  = ws + BCE_OFF;   // 4096 floats
    float* distP = ws + DIST_OFF;  // 33792 floats

    // 8 waves (rows) per 256-thread block -> 8192 blocks cover 65536 rows
    mk_norms<<<NNORMS / 8, 256, 0, stream>>>(x, norms);
    mk_bce<<<BCE_BLOCKS, 256, 0, stream>>>(x, t, bceP);
    mk_dist<<<DIST_BLOCKS, 32, 0, stream>>>(x, norms, distP);
    mk_finish<<<1, 256, 0, stream>>>(bceP, distP, out);
}